// SAGEConv_15324443312418
// MI455X (gfx1250) — compile-verified
//
#include <hip/hip_runtime.h>
#include <hip/hip_bf16.h>

// ---------------------------------------------------------------------------
// SAGEConv: m = h[src]*w ; agg = mean-by-dst(m) ; out = relu([h|agg] @ W^T + b)
// N=100000 nodes, E=1250000 edges, F_in=64, F_out=64, K=128
// ---------------------------------------------------------------------------

typedef float v2f __attribute__((ext_vector_type(2)));
typedef float v8f __attribute__((ext_vector_type(8)));

#define N_FEAT 64
#define K_TOT  128          // 2*N_FEAT
#define TILE_M 64           // nodes per block (4 waves x 16 nodes)
#define LDS_STRIDE 130      // 128 + 2 pad dwords -> conflict-free ds_load_b64

// ---------------------------------------------------------------------------
// Kernel 0: zero the workspace (agg[N*64] ++ cnt[N])
// ---------------------------------------------------------------------------
__global__ void sage_zero_kernel(float* p, int n) {
    int i = blockIdx.x * blockDim.x + threadIdx.x;
    if (i < n) p[i] = 0.0f;
}

// ---------------------------------------------------------------------------
// Kernel 1: edge scatter.  16 lanes per edge, one float4 of features per lane.
// agg (25.6MB) lives in L2 (192MB) -> native global_atomic_add_f32 is cheap.
// ---------------------------------------------------------------------------
__global__ __launch_bounds__(256)
void sage_edge_kernel(const float* __restrict__ h,
                      const float* __restrict__ w,
                      const int*   __restrict__ src,
                      const int*   __restrict__ dst,
                      float* __restrict__ agg,
                      float* __restrict__ cnt,
                      int n_edges) {
    int tid  = blockIdx.x * blockDim.x + threadIdx.x;
    int e    = tid >> 4;          // edge index
    int sub  = tid & 15;          // feature chunk (float4)
    if (e >= n_edges) return;

    int s  = src[e];
    int d  = dst[e];
    float we = w[e];

    const float4 hv = *(const float4*)(h + (size_t)s * N_FEAT + sub * 4);
    float* ap = agg + (size_t)d * N_FEAT + sub * 4;
    unsafeAtomicAdd(ap + 0, hv.x * we);
    unsafeAtomicAdd(ap + 1, hv.y * we);
    unsafeAtomicAdd(ap + 2, hv.z * we);
    unsafeAtomicAdd(ap + 3, hv.w * we);
    if (sub == 0) unsafeAtomicAdd(cnt + d, 1.0f);
}

// ---------------------------------------------------------------------------
// Kernel 2: fused normalize + concat + GEMM + bias + relu using
// V_WMMA_F32_16X16X4_F32.  Each wave: 16 nodes x 64 outputs, K=128.
//
// f32 WMMA VGPR layouts (ISA 7.12.2):
//   A 16x4 : lane L holds A[M=L%16][K=(L/16)*2 + v], v=0..1  -> v2f
//   B 4x16 : lane L holds B[K=(L/16)*2 + v][N=L%16]          -> v2f
//   C/D    : lane L, reg r holds D[M=r+8*(L/16)][N=L%16]     -> v8f
// ---------------------------------------------------------------------------
__global__ __launch_bounds__(128)
void sage_gemm_kernel(const float* __restrict__ h,
                      const float* __restrict__ agg,
                      const float* __restrict__ cnt,
                      const float* __restrict__ Wm,   // [64,128] row-major
                      const float* __restrict__ bias, // [64]
                      float* __restrict__ out,        // [N,64]
                      int n_nodes) {
    __shared__ float sH[TILE_M * LDS_STRIDE];  // 64 x 128 h_total tile (padded)
    __shared__ float sInv[TILE_M];

    const int base = blockIdx.x * TILE_M;

    // --- in-degree normalizers for this tile ---
    for (int i = threadIdx.x; i < TILE_M; i += blockDim.x) {
        int node = min(base + i, n_nodes - 1);
        sInv[i] = 1.0f / fmaxf(cnt[node], 1.0f);
    }
    __syncthreads();

    // --- stage h_total = [h | agg/max(cnt,1)] into LDS (float2 granularity,
    //     8B-aligned since LDS_STRIDE is even) ---
    for (int i = threadIdx.x; i < TILE_M * (K_TOT / 2); i += blockDim.x) {
        int row = i >> 6;               // / (K_TOT/2)
        int col = (i & 63) * 2;
        int node = min(base + row, n_nodes - 1);
        float2 v;
        if (col < N_FEAT) {
            v = *(const float2*)(h + (size_t)node * N_FEAT + col);
        } else {
            float2 a = *(const float2*)(agg + (size_t)node * N_FEAT + (col - N_FEAT));
            float s = sInv[row];
            v = make_float2(a.x * s, a.y * s);
        }
        *(float2*)(&sH[row * LDS_STRIDE + col]) = v;
    }
    __syncthreads();

    // --- per-wave WMMA tile ---
    const int wv   = threadIdx.x >> 5;   // 0..3  -> node sub-tile
    const int lane = threadIdx.x & 31;
    const int mrow = lane & 15;          // M (for A) / N (for B,C,D)
    const int half = lane >> 4;          // 0/1
    const int kk   = half * 2;           // K sub-offset within each K=4 step

    const float* aBase = &sH[(wv * 16 + mrow) * LDS_STRIDE];
    const float* w0 = Wm + (size_t)(0 * 16 + mrow) * K_TOT;
    const float* w1 = Wm + (size_t)(1 * 16 + mrow) * K_TOT;
    const float* w2 = Wm + (size_t)(2 * 16 + mrow) * K_TOT;
    const float* w3 = Wm + (size_t)(3 * 16 + mrow) * K_TOT;

    v8f acc0 = {}, acc1 = {}, acc2 = {}, acc3 = {};

    #pragma unroll 4
    for (int t = 0; t < K_TOT / 4; ++t) {
        const int k = 4 * t + kk;
        v2f a  = *(const v2f*)(aBase + k);   // ds_load_b64, conflict-free
        v2f b0 = *(const v2f*)(w0 + k);      // global, resident in WGP$/L2
        v2f b1 = *(const v2f*)(w1 + k);
        v2f b2 = *(const v2f*)(w2 + k);
        v2f b3 = *(const v2f*)(w3 + k);
        acc0 = __builtin_amdgcn_wmma_f32_16x16x4_f32(false, a, false, b0, (short)0, acc0, false, false);
        acc1 = __builtin_amdgcn_wmma_f32_16x16x4_f32(false, a, false, b1, (short)0, acc1, false, false);
        acc2 = __builtin_amdgcn_wmma_f32_16x16x4_f32(false, a, false, b2, (short)0, acc2, false, false);
        acc3 = __builtin_amdgcn_wmma_f32_16x16x4_f32(false, a, false, b3, (short)0, acc3, false, false);
    }

    const float bn0 = bias[0 * 16 + mrow];
    const float bn1 = bias[1 * 16 + mrow];
    const float bn2 = bias[2 * 16 + mrow];
    const float bn3 = bias[3 * 16 + mrow];

    #pragma unroll
    for (int r = 0; r < 8; ++r) {
        int node = base + wv * 16 + half * 8 + r;
        if (node < n_nodes) {
            float* o = out + (size_t)node * N_FEAT;
            o[0 * 16 + mrow] = fmaxf(acc0[r] + bn0, 0.0f);
            o[1 * 16 + mrow] = fmaxf(acc1[r] + bn1, 0.0f);
            o[2 * 16 + mrow] = fmaxf(acc2[r] + bn2, 0.0f);
            o[3 * 16 + mrow] = fmaxf(acc3[r] + bn3, 0.0f);
        }
    }
}

// ---------------------------------------------------------------------------
extern "C" void kernel_launch(void* const* d_in, const int* in_sizes, int n_in,
                              void* d_out, int out_size, void* d_ws, size_t ws_size,
                              hipStream_t stream) {
    const float* h   = (const float*)d_in[0];
    const float* w   = (const float*)d_in[1];
    const int*   src = (const int*)  d_in[2];
    const int*   dst = (const int*)  d_in[3];
    const float* Wm  = (const float*)d_in[4];
    const float* b   = (const float*)d_in[5];
    float*       out = (float*)d_out;

    const int n_nodes = in_sizes[0] / N_FEAT;
    const int n_edges = in_sizes[1];

    float* agg = (float*)d_ws;                       // [N,64]
    float* cnt = agg + (size_t)n_nodes * N_FEAT;     // [N]

    // zero workspace (harness does not re-zero between replays)
    {
        int z = n_nodes * N_FEAT + n_nodes;
        int blk = (z + 255) / 256;
        sage_zero_kernel<<<blk, 256, 0, stream>>>(agg, z);
    }
    // edge scatter: 16 lanes per edge
    {
        long long threads = (long long)n_edges * 16;
        int blk = (int)((threads + 255) / 256);
        sage_edge_kernel<<<blk, 256, 0, stream>>>(h, w, src, dst, agg, cnt, n_edges);
    }
    // fused normalize + GEMM (f32 WMMA) + bias + relu
    {
        int blk = (n_nodes + TILE_M - 1) / TILE_M;
        sage_gemm_kernel<<<blk, 128, 0, stream>>>(h, agg, cnt, Wm, b, out, n_nodes);
    }
}